// GradientPerturbationController_11149735101053
// MI455X (gfx1250) — compile-verified
//
#include <hip/hip_runtime.h>
#include <math.h>

#define NN   128          // state dim
#define MCC  64           // control dim
#define TT   200          // timesteps
#define HHOR 5            // horizon h
#define MW   5            // mwin
#define DD   (NN*MW)      // 640 flattened (j,i) dim of E
#define WCOLS 18          // padded row stride of Wmat in LDS (even -> 8B-aligned float2)
#define ETA_C  1e-3f
#define WD_C   1e-5f
#define B1_C   0.9f
#define B2_C   0.999f
#define EPS_C  1e-8f
#define MAXN_C 5.0f

typedef float v2f __attribute__((ext_vector_type(2)));
typedef float v8f __attribute__((ext_vector_type(8)));

// Single persistent workgroup runs all 200 sequential timesteps.
// WMMA use: V_WMMA_F32_16X16X4_F32 for (a) Vpert = E(64x640) @ Wmat(640x16)
// and (b) gE = Gv(64x8) @ Wmat^T(8x640), Adam fused into the gE tile scatter.
__global__ __launch_bounds__(256)
void gpc_persistent_kernel(const float* __restrict__ Ag, const float* __restrict__ Bg,
                           const float* __restrict__ Qg, const float* __restrict__ Rg,
                           const float* __restrict__ Kg, const float* __restrict__ Wg,
                           const float* __restrict__ E0, const float* __restrict__ b0,
                           float* __restrict__ losses,
                           float* __restrict__ Eg, float* __restrict__ mEg, float* __restrict__ vEg)
{
    const int tid  = threadIdx.x;                                  // 256 threads = 8 waves
    const int lane = tid & 31;
    const int wave = __builtin_amdgcn_readfirstlane(tid >> 5);     // scalar -> uniform branch, EXEC stays full for WMMA

    extern __shared__ float smem[];
    float* Wm  = smem;              // [640][18] sliding-window matrix (cols 0..5 live, rest 0)
    float* Vp  = Wm  + DD*WCOLS;    // [64][6]  perturbation columns (hh=0..4) + outer u_pert (col 5)
    float* Gv  = Vp  + MCC*6;       // [64][8]  per-horizon v-gradients, K-padded with zeros
    float* wh  = Gv  + MCC*8;       // [10][128] circular w history; slot t%10 holds W[t]
    float* ys  = wh  + 10*NN;       // [6][128] proxy rollout states
    float* vs  = ys  + 6*NN;        // [5][64]  proxy rollout controls
    float* xv  = vs  + HHOR*MCC;    // [128] x
    float* uv  = xv  + NN;          // [64]  u
    float* zv  = uv  + MCC;         // [128] scratch for dyn
    float* gy  = zv  + NN;          // [128] adjoint carry
    float* gyf = gy  + NN;          // [128] adjoint incl. cost term
    float* gvv = gyf + NN;          // [64]  gv
    float* bia = gvv + MCC;         // [64]  bias param
    float* mb  = bia + MCC;         // [64]  Adam m(bias)
    float* vb  = mb  + MCC;         // [64]  Adam v(bias)
    float* red = vb  + MCC;         // [256] reduction scratch
    // total = 15872 floats = 63488 bytes

    // ---- init (deterministic every launch; ws is poisoned by harness) ----
    for (int i = tid; i < MCC*DD; i += 256) { Eg[i] = E0[i]; mEg[i] = 0.f; vEg[i] = 0.f; }
    for (int i = tid; i < 10*NN;  i += 256) wh[i] = 0.f;
    if (tid < NN) { xv[tid] = 0.f; zv[tid] = 0.f; gy[tid] = 0.f; }
    if (tid < MCC){ uv[tid] = 0.f; bia[tid] = b0[tid]; mb[tid] = 0.f; vb[tid] = 0.f; }
    __syncthreads();

    for (int t = 0; t < TT; ++t) {
        // ---- region 1: dyn = A x + B u + w_t ; push w_t into history ----
        if (tid < NN) {
            float s = Wg[t*NN + tid];
            const float* ar = Ag + tid*NN;
            for (int c = 0; c < NN; ++c)  s += ar[c] * xv[c];
            const float* br = Bg + tid*MCC;
            for (int c = 0; c < MCC; ++c) s += br[c] * uv[c];
            zv[tid] = (t == 0) ? 0.f : s;
            wh[(t % 10)*NN + tid] = Wg[t*NN + tid];
        }
        __syncthreads();

        // ---- region 2: commit x; build Wmat[d][col] = w_hist[col + d%5][d/5] ----
        if (tid < NN) xv[tid] = zv[tid];
        {
            const int base = (t + 1) % 10;          // w_hist[k] -> wh slot (base+k)%10
            for (int i = tid; i < DD*WCOLS; i += 256) {
                int d = i / WCOLS, col = i % WCOLS;
                float val = 0.f;
                if (col < 6) {
                    int slot = (base + col + (d % MW)) % 10;
                    val = wh[slot*NN + (d / MW)];
                }
                Wm[i] = val;
            }
        }
        __syncthreads();

        // ---- region 3: Vpert(64x16) = E(64x640) @ Wmat ; waves 0..3, one 16-row tile each ----
        if (wave < 4) {
            const int mt    = wave;
            const int m     = mt*16 + (lane & 15);
            const int khalf = 2*(lane >> 4);
            const int n     = lane & 15;
            v8f acc = {0.f,0.f,0.f,0.f,0.f,0.f,0.f,0.f};
            const float* erow = Eg + m*DD;
            for (int ks = 0; ks < DD/4; ++ks) {
                const int kb = ks*4 + khalf;                    // even -> 8B-aligned
                v2f a = *(const v2f*)(erow + kb);               // A frag: E[m, kb..kb+1]
                v2f b;                                          // B frag: Wmat[kb..kb+1][n]
                b.x = Wm[kb*WCOLS + n];
                b.y = Wm[(kb+1)*WCOLS + n];
                acc = __builtin_amdgcn_wmma_f32_16x16x4_f32(false, a, false, b,
                                                            (short)0, acc, false, false);
            }
            const int rbase = mt*16 + 8*(lane >> 4);
            #pragma unroll
            for (int v = 0; v < 8; ++v)
                if (n < 6) Vp[(rbase + v)*6 + n] = acc[v];
        }
        __syncthreads();

        // ---- region 4: u = -K x + Vpert[:,5] + bias ----
        if (tid < MCC) {
            float s = Vp[tid*6 + 5] + bia[tid];
            const float* kr = Kg + tid*NN;
            for (int c = 0; c < NN; ++c) s -= kr[c] * xv[c];
            uv[tid] = s;
        }
        __syncthreads();

        // ---- region 5: loss = x'Qx + u'Ru ----
        {
            float p = 0.f;
            if (tid < NN) {
                float s = 0.f;
                const float* qr = Qg + tid*NN;
                for (int c = 0; c < NN; ++c) s += qr[c] * xv[c];
                p = xv[tid] * s;
            }
            if (tid < MCC) {
                float s = 0.f;
                const float* rr = Rg + tid*MCC;
                for (int c = 0; c < MCC; ++c) s += rr[c] * uv[c];
                p += uv[tid] * s;
            }
            red[tid] = p;
            __syncthreads();
            for (int s2 = 128; s2 > 0; s2 >>= 1) {
                if (tid < s2) red[tid] += red[tid + s2];
                __syncthreads();
            }
            if (tid == 0) losses[t] = red[0];
            __syncthreads();
        }

        // ---- region 6: proxy gradient + Adam updates (only when epoch >= 0) ----
        if (t >= HHOR + MW) {
            const int base = (t + 1) % 10;
            // forward rollout (y0 = 0), Vpert columns 0..4 already hold the einsum terms
            if (tid < NN) ys[tid] = 0.f;
            __syncthreads();
            for (int hh = 0; hh < HHOR; ++hh) {
                if (tid < MCC) {
                    float s = Vp[tid*6 + hh] + bia[tid];
                    const float* kr = Kg + tid*NN;
                    const float* yc = ys + hh*NN;
                    for (int c = 0; c < NN; ++c) s -= kr[c] * yc[c];
                    vs[hh*MCC + tid] = s;
                }
                __syncthreads();
                if (tid < NN) {
                    float s = wh[((base + hh + MW) % 10)*NN + tid];
                    const float* ar = Ag + tid*NN;
                    const float* yc = ys + hh*NN;
                    for (int c = 0; c < NN; ++c)  s += ar[c] * yc[c];
                    const float* br = Bg + tid*MCC;
                    const float* vc = vs + hh*MCC;
                    for (int c = 0; c < MCC; ++c) s += br[c] * vc[c];
                    ys[(hh+1)*NN + tid] = s;
                }
                __syncthreads();
            }
            // reverse sweep: gv_hh = B' gyf + (R+R') v ; gy = A' gyf - K' gv
            for (int i = tid; i < MCC*8; i += 256) Gv[i] = 0.f;   // K-padding zeros (cols 5..7)
            if (tid < NN) gy[tid] = 0.f;
            __syncthreads();
            for (int hh = HHOR - 1; hh >= 0; --hh) {
                if (tid < NN) {
                    float s = gy[tid];
                    const float* yc = ys + (hh+1)*NN;
                    for (int c = 0; c < NN; ++c)
                        s += (Qg[tid*NN + c] + Qg[c*NN + tid]) * yc[c];
                    gyf[tid] = s;
                }
                __syncthreads();
                if (tid < MCC) {
                    float s = 0.f;
                    for (int c = 0; c < NN; ++c)  s += Bg[c*MCC + tid] * gyf[c];
                    const float* vc = vs + hh*MCC;
                    for (int c = 0; c < MCC; ++c) s += (Rg[tid*MCC + c] + Rg[c*MCC + tid]) * vc[c];
                    gvv[tid] = s;
                    Gv[tid*8 + hh] = s;
                }
                __syncthreads();
                if (tid < NN) {
                    float s = 0.f;
                    for (int c = 0; c < NN; ++c)  s += Ag[c*NN + tid] * gyf[c];
                    for (int c = 0; c < MCC; ++c) s -= Kg[c*NN + tid] * gvv[c];
                    gy[tid] = s;
                }
                __syncthreads();
            }

            // Adam scalars (uniform)
            const float ef     = (float)(t - (HHOR + MW));
            const float factor = (ef < 10.f) ? 0.1f
                                 : ((ef < 50.f) ? 1.f : fmaxf(0.1f, 1.f - (ef - 50.f)/100.f));
            const float lr     = ETA_C * factor;
            const float stepf  = ef + 1.f;
            const float bc1    = 1.f - powf(B1_C, stepf);
            const float bc2    = 1.f - powf(B2_C, stepf);

            // gE = Gv @ Wmat^T via WMMA, Adam fused on the output tiles (all 8 waves)
            float nrm2 = 0.f;
            {
                const int khalf = 2*(lane >> 4);
                const int lm    = lane & 15;
                for (int j = wave; j < 4*40; j += 8) {           // 160 tiles, 20 per wave
                    const int mt = j / 40, nt = j % 40;
                    const int m  = mt*16 + lm;
                    v8f acc = {0.f,0.f,0.f,0.f,0.f,0.f,0.f,0.f};
                    #pragma unroll
                    for (int ks = 0; ks < 2; ++ks) {
                        const int kb = ks*4 + khalf;
                        v2f a; a.x = Gv[m*8 + kb]; a.y = Gv[m*8 + kb + 1];
                        v2f b = *(const v2f*)(Wm + (nt*16 + lm)*WCOLS + kb); // Wmat[n][kb..kb+1]
                        acc = __builtin_amdgcn_wmma_f32_16x16x4_f32(false, a, false, b,
                                                                    (short)0, acc, false, false);
                    }
                    const int rbase = mt*16 + 8*(lane >> 4);
                    const int dcol  = nt*16 + lm;
                    #pragma unroll
                    for (int v = 0; v < 8; ++v) {
                        const int idx = (rbase + v)*DD + dcol;
                        float p  = Eg[idx];
                        float g  = acc[v] + WD_C * p;
                        float m1 = B1_C*mEg[idx] + (1.f - B1_C)*g;
                        float v1 = B2_C*vEg[idx] + (1.f - B2_C)*g*g;
                        float p1 = p - lr * (m1/bc1) / (sqrtf(v1/bc2) + EPS_C);
                        mEg[idx] = m1; vEg[idx] = v1; Eg[idx] = p1;
                        nrm2 += p1*p1;
                    }
                }
            }
            red[tid] = nrm2;
            __syncthreads();
            for (int s2 = 128; s2 > 0; s2 >>= 1) {
                if (tid < s2) red[tid] += red[tid + s2];
                __syncthreads();
            }
            const float nrm = sqrtf(red[0]);
            __syncthreads();
            if (nrm > MAXN_C) {
                const float sc = MAXN_C / nrm;
                for (int i = tid; i < MCC*DD; i += 256) Eg[i] *= sc;
            }
            // bias Adam (no clip)
            if (tid < MCC) {
                float g = 0.f;
                for (int h2 = 0; h2 < HHOR; ++h2) g += Gv[tid*8 + h2];
                g += WD_C * bia[tid];
                float m1 = B1_C*mb[tid] + (1.f - B1_C)*g;
                float v1 = B2_C*vb[tid] + (1.f - B2_C)*g*g;
                bia[tid] -= lr * (m1/bc1) / (sqrtf(v1/bc2) + EPS_C);
                mb[tid] = m1; vb[tid] = v1;
            }
            __syncthreads();
        }
    }
}

extern "C" void kernel_launch(void* const* d_in, const int* in_sizes, int n_in,
                              void* d_out, int out_size, void* d_ws, size_t ws_size,
                              hipStream_t stream) {
    const float* Ag = (const float*)d_in[0];   // 128x128
    const float* Bg = (const float*)d_in[1];   // 128x64
    const float* Qg = (const float*)d_in[2];   // 128x128
    const float* Rg = (const float*)d_in[3];   // 64x64
    const float* Kg = (const float*)d_in[4];   // 64x128
    const float* Wg = (const float*)d_in[5];   // 200x128x1
    const float* E0 = (const float*)d_in[6];   // 64x128x5
    const float* b0 = (const float*)d_in[7];   // 64x1
    // d_in[8]=h, d_in[9]=mwin (hardcoded 5/5)
    float* losses = (float*)d_out;             // 200 f32
    float* Eg  = (float*)d_ws;                 // 40960 f32
    float* mEg = Eg  + MCC*DD;                 // 40960 f32
    float* vEg = mEg + MCC*DD;                 // 40960 f32

    const size_t smem_bytes = 15872 * sizeof(float);   // 63488 B dynamic LDS
    gpc_persistent_kernel<<<1, 256, smem_bytes, stream>>>(
        Ag, Bg, Qg, Rg, Kg, Wg, E0, b0, losses, Eg, mEg, vEg);
}